// GCN2_52364241273198
// MI455X (gfx1250) — compile-verified
//
#include <hip/hip_runtime.h>
#include <math.h>

#define NN     100000
#define EE     1600000
#define FIN    300
#define HID    64
#define OUTC   20
#define NLAYER 8
#define ALPHA  0.1f

typedef float v2f __attribute__((ext_vector_type(2)));
typedef float v8f __attribute__((ext_vector_type(8)));

__device__ __forceinline__ v8f wmma_f32_16x16x4(v2f a, v2f b, v8f c) {
    // exact fp32 matrix op: V_WMMA_F32_16X16X4_F32
    return __builtin_amdgcn_wmma_f32_16x16x4_f32(false, a, false, b, (short)0, c,
                                                 false, false);
}

// ---------------------------------------------------------------- utilities
__global__ void k_fill(float* __restrict__ p, float v, int n) {
    int i = blockIdx.x * blockDim.x + threadIdx.x;
    if (i < n) p[i] = v;
}

__global__ void k_deg_acc(const int* __restrict__ col, const float* __restrict__ w,
                          float* __restrict__ deg) {
    int e = blockIdx.x * blockDim.x + threadIdx.x;
    if (e < EE) atomicAdd(&deg[col[e]], w[e]);
}

__global__ void k_dinv(float* __restrict__ deg) {
    int i = blockIdx.x * blockDim.x + threadIdx.x;
    if (i < NN) {
        float d = deg[i];
        deg[i] = d > 0.f ? rsqrtf(d) : 0.f;
    }
}

__global__ void k_norm(const int* __restrict__ row, const int* __restrict__ col,
                       const float* __restrict__ w, const float* __restrict__ dinv,
                       float* __restrict__ norm) {
    int e = blockIdx.x * blockDim.x + threadIdx.x;
    if (e < EE) norm[e] = dinv[row[e]] * w[e] * dinv[col[e]];
}

// ------------------------------------------------- input GEMM: h = relu(x@Win + b)
// grid = NN/16 blocks, 128 threads (4 waves); wave w does N-tile [16w,16w+16)
__global__ __launch_bounds__(128)
void k_gemm_in(const float* __restrict__ x, const float* __restrict__ Win,
               const float* __restrict__ bin, float* __restrict__ x0,
               float* __restrict__ h) {
    __shared__ float xs[16][FIN + 4];   // pad to 304 floats/row
    const int tid = threadIdx.x;
    const int mb  = blockIdx.x;

    for (int i = tid; i < 16 * FIN; i += 128) {
        int r = i / FIN, c = i - r * FIN;
        xs[r][c] = x[(size_t)(mb * 16 + r) * FIN + c];
    }
    __syncthreads();

    const int lane = tid & 31;
    const int n0   = (tid >> 5) * 16;
    const int m    = lane & 15;          // A: row
    const int kh   = (lane >> 4) * 2;    // A/B: K sub-offset
    const int nl   = lane & 15;          // B/D: col

    v8f acc = {};
    for (int k0 = 0; k0 < FIN; k0 += 4) {
        v2f a, b;
        a[0] = xs[m][k0 + kh];
        a[1] = xs[m][k0 + kh + 1];
        b[0] = Win[(size_t)(k0 + kh)     * HID + n0 + nl];
        b[1] = Win[(size_t)(k0 + kh + 1) * HID + n0 + nl];
        acc  = wmma_f32_16x16x4(a, b, acc);
    }

    #pragma unroll
    for (int v = 0; v < 8; ++v) {
        int mm  = v + (lane >> 4) * 8;
        int c   = n0 + nl;
        float val = acc[v] + bin[c];
        val = val > 0.f ? val : 0.f;
        size_t idx = (size_t)(mb * 16 + mm) * HID + c;
        x0[idx] = val;
        h[idx]  = val;
    }
}

// ---------------------------------------------- SpMM scatter: agg += norm * h[row]
// one wave per edge; lane handles 2 consecutive features (64 = 32 lanes * 2)
__global__ __launch_bounds__(256)
void k_scatter(const int* __restrict__ row, const int* __restrict__ col,
               const float* __restrict__ norm, const float* __restrict__ h,
               float* __restrict__ agg) {
    int gid  = blockIdx.x * blockDim.x + threadIdx.x;
    int lane = gid & 31;
    int wid  = gid >> 5;
    int nwv  = (gridDim.x * blockDim.x) >> 5;
    for (int e = wid; e < EE; e += nwv) {
        int   r  = row[e];
        int   c  = col[e];
        float nv = norm[e];
        float2 hv = *reinterpret_cast<const float2*>(h + (size_t)r * HID + lane * 2);
        atomicAdd(agg + (size_t)c * HID + lane * 2 + 0, nv * hv.x);
        atomicAdd(agg + (size_t)c * HID + lane * 2 + 1, nv * hv.y);
    }
}

// ---- fused layer: z = (1-a)*(agg + dinv^2*h) + a*x0 ; h = relu((1-b)z + b*(z@Wl))
__global__ __launch_bounds__(128)
void k_layer(const float* __restrict__ agg, const float* __restrict__ x0,
             const float* __restrict__ dinv, const float* __restrict__ Wl,
             float* __restrict__ h, float beta) {
    __shared__ float zs[16][HID + 8];    // 16x72, avoids bank conflicts
    __shared__ float ws[HID][HID];       // 64x64 layer weight, 16 KB
    const int tid = threadIdx.x;
    const int mb  = blockIdx.x;

    for (int i = tid; i < 16 * HID; i += 128) {
        int r = i >> 6, c = i & 63;
        int rowi = mb * 16 + r;
        float di = dinv[rowi];
        size_t idx = (size_t)rowi * HID + c;
        float a = agg[idx] + di * di * h[idx];   // add implicit self-loop
        zs[r][c] = (1.f - ALPHA) * a + ALPHA * x0[idx];
    }
    for (int i = tid; i < HID * HID; i += 128)
        ws[i >> 6][i & 63] = Wl[i];
    __syncthreads();

    const int lane = tid & 31;
    const int n0   = (tid >> 5) * 16;
    const int m    = lane & 15;
    const int kh   = (lane >> 4) * 2;
    const int nl   = lane & 15;

    v8f acc = {};
    #pragma unroll
    for (int k0 = 0; k0 < HID; k0 += 4) {
        v2f a, b;
        a[0] = zs[m][k0 + kh];
        a[1] = zs[m][k0 + kh + 1];
        b[0] = ws[k0 + kh][n0 + nl];
        b[1] = ws[k0 + kh + 1][n0 + nl];
        acc  = wmma_f32_16x16x4(a, b, acc);
    }

    #pragma unroll
    for (int v = 0; v < 8; ++v) {
        int mm = v + (lane >> 4) * 8;
        float z   = zs[mm][n0 + nl];
        float val = (1.f - beta) * z + beta * acc[v];
        val = val > 0.f ? val : 0.f;
        h[(size_t)(mb * 16 + mm) * HID + n0 + nl] = val;
    }
}

// ------------------------------------------------ output: out = h@Wout + bout
__global__ void k_out(const float* __restrict__ h, const float* __restrict__ Wout,
                      const float* __restrict__ bout, float* __restrict__ out) {
    int i = blockIdx.x * blockDim.x + threadIdx.x;
    if (i >= NN * OUTC) return;
    int n = i / OUTC, c = i - n * OUTC;
    float s = bout[c];
    const float* hr = h + (size_t)n * HID;
    #pragma unroll 16
    for (int k = 0; k < HID; ++k) s += hr[k] * Wout[k * OUTC + c];
    out[i] = s;
}

// ---------------------------------------------------------------- launcher
extern "C" void kernel_launch(void* const* d_in, const int* in_sizes, int n_in,
                              void* d_out, int out_size, void* d_ws, size_t ws_size,
                              hipStream_t stream) {
    (void)in_sizes; (void)n_in; (void)out_size; (void)ws_size;

    const float* x     = (const float*)d_in[0];
    const int*   ei    = (const int*)  d_in[1];
    const float* ew    = (const float*)d_in[2];
    const float* Win   = (const float*)d_in[3];
    const float* bin   = (const float*)d_in[4];
    const float* convw = (const float*)d_in[5];
    const float* Wout  = (const float*)d_in[6];
    const float* bout  = (const float*)d_in[7];
    const int* row = ei;        // edge_index[0]
    const int* col = ei + EE;   // edge_index[1]

    char*  ws  = (char*)d_ws;
    size_t off = 0;
    auto carve = [&](size_t bytes) -> float* {
        float* p = (float*)(ws + off);
        off = (off + bytes + 255) & ~(size_t)255;
        return p;
    };
    float* dinv = carve((size_t)NN * 4);          // degrees -> dinv (in place)
    float* norm = carve((size_t)EE * 4);
    float* x0   = carve((size_t)NN * HID * 4);
    float* h    = carve((size_t)NN * HID * 4);
    float* agg  = carve((size_t)NN * HID * 4);

    // gcn_norm: deg starts at 1 (self loop weight), accumulate edge weights on col
    k_fill   <<<(NN + 255) / 256, 256, 0, stream>>>(dinv, 1.0f, NN);
    k_deg_acc<<<(EE + 255) / 256, 256, 0, stream>>>(col, ew, dinv);
    k_dinv   <<<(NN + 255) / 256, 256, 0, stream>>>(dinv);
    k_norm   <<<(EE + 255) / 256, 256, 0, stream>>>(row, col, ew, dinv, norm);

    // h = x0 = relu(x @ W_in + b_in)
    k_gemm_in<<<NN / 16, 128, 0, stream>>>(x, Win, bin, x0, h);

    for (int l = 0; l < NLAYER; ++l) {
        float beta = logf(0.5f / (float)(l + 1) + 1.0f);
        k_fill   <<<(NN * HID + 255) / 256, 256, 0, stream>>>(agg, 0.0f, NN * HID);
        k_scatter<<<2048, 256, 0, stream>>>(row, col, norm, h, agg);
        k_layer  <<<NN / 16, 128, 0, stream>>>(agg, x0, dinv,
                                               convw + (size_t)l * HID * HID, h, beta);
    }

    k_out<<<(NN * OUTC + 255) / 256, 256, 0, stream>>>(h, Wout, bout, (float*)d_out);
}